// Predict_modelv3_65704409694778
// MI455X (gfx1250) — compile-verified
//
#include <hip/hip_runtime.h>
#include <hip/hip_bf16.h>
#include <stdint.h>

#define EMB    128
#define HEADS  4
#define HC     (HEADS * EMB)      // 512
#define LAYERS 2
#define T_WIN  10
#define NNODES 20000
#define NEDGES 320000
#define ETOT   (NEDGES + NNODES)  // 340000, with self loops
#define BCLF   4096
#define FLATW  (T_WIN * 2 * EMB)  // 2560

typedef __attribute__((ext_vector_type(2))) float v2f;
typedef __attribute__((ext_vector_type(8))) float v8f;

#define WMMA_F32(A_, B_, C_) \
    __builtin_amdgcn_wmma_f32_16x16x4_f32(false, (A_), false, (B_), (short)0, (C_), false, false)

// ---------------------------------------------------------------------------
// fp32 WMMA GEMM: C[M,Nc] = act( A[M,K] * Wt[Nc,K]^T + bias )
// One wave -> 32x64 tile: 2 A fragments x 4 W fragments -> 8 independent
// accumulators (6 loads feed 8 WMMAs, ~10.7 FLOP/B from L2).
// Fragment loads are software-pipelined one k-step ahead so WMMAs consume
// *older* loads (partial s_wait_loadcnt instead of full serialization).
// 4 waves/block cover 128 rows; partial M handled by wave-uniform exit.
// ---------------------------------------------------------------------------
__global__ void gemm_wmma_f32(const float* __restrict__ A,
                              const float* __restrict__ Wt,
                              const float* __restrict__ bias,
                              float* __restrict__ C,
                              int M, int Nc, int K, int relu) {
    const int wave = threadIdx.x >> 5;
    const int lane = threadIdx.x & 31;
    const int half = lane >> 4;        // 0: K pair {0,1} / M rows 0..7 ; 1: {2,3} / 8..15
    const int l    = lane & 15;

    const int m0 = blockIdx.y * 128 + wave * 32;
    if (m0 >= M) return;                           // wave-uniform (EXEC all-ones below)
    const int n0 = blockIdx.x * 64;

    const float* ap = A  + (size_t)(m0 + l) * K + 2 * half;   // +16*K for 2nd m-tile
    const float* wp = Wt + (size_t)(n0 + l) * K + 2 * half;   // +16*K*j for j-th n-tile

    v8f acc[2][4] = {};
    v2f af[2], wf[4], naf[2], nwf[4];

    af[0] = *(const v2f*)(ap);
    af[1] = *(const v2f*)(ap + (size_t)16 * K);
    wf[0] = *(const v2f*)(wp);
    wf[1] = *(const v2f*)(wp + (size_t)16 * K);
    wf[2] = *(const v2f*)(wp + (size_t)32 * K);
    wf[3] = *(const v2f*)(wp + (size_t)48 * K);

    for (int k0 = 4; k0 < K; k0 += 4) {
        // prefetch next k-step (stays in flight across the WMMAs below)
        naf[0] = *(const v2f*)(ap + k0);
        naf[1] = *(const v2f*)(ap + (size_t)16 * K + k0);
        nwf[0] = *(const v2f*)(wp + k0);
        nwf[1] = *(const v2f*)(wp + (size_t)16 * K + k0);
        nwf[2] = *(const v2f*)(wp + (size_t)32 * K + k0);
        nwf[3] = *(const v2f*)(wp + (size_t)48 * K + k0);

        #pragma unroll
        for (int i = 0; i < 2; ++i)
            #pragma unroll
            for (int j = 0; j < 4; ++j)
                acc[i][j] = WMMA_F32(af[i], wf[j], acc[i][j]);

        af[0] = naf[0]; af[1] = naf[1];
        wf[0] = nwf[0]; wf[1] = nwf[1]; wf[2] = nwf[2]; wf[3] = nwf[3];
    }
    #pragma unroll
    for (int i = 0; i < 2; ++i)
        #pragma unroll
        for (int j = 0; j < 4; ++j)
            acc[i][j] = WMMA_F32(af[i], wf[j], acc[i][j]);

    #pragma unroll
    for (int mi = 0; mi < 2; ++mi) {
        float* cr = C + (size_t)(m0 + 16 * mi + 8 * half) * Nc + n0 + l;
        #pragma unroll
        for (int j = 0; j < 4; ++j) {
            const float bv = bias ? bias[n0 + 16 * j + l] : 0.0f;
            #pragma unroll
            for (int r = 0; r < 8; ++r) {
                float v = acc[mi][j][r] + bv;
                if (relu) v = fmaxf(v, 0.0f);
                cr[(size_t)r * Nc + 16 * j] = v;
            }
        }
    }
}

// ---------------------------------------------------------------------------
// Per-node attention coefficients: asn[n,h] = <h[n,h,:], a_s[h,:]> (same for adn)
// One wave per (n, head).
// ---------------------------------------------------------------------------
__global__ void attn_coeff(const float* __restrict__ h,
                           const float* __restrict__ a_s,
                           const float* __restrict__ a_d,
                           float* __restrict__ asn, float* __restrict__ adn) {
    int gw = blockIdx.x * (blockDim.x >> 5) + (threadIdx.x >> 5);
    if (gw >= NNODES * HEADS) return;
    int n = gw >> 2, hd = gw & 3;
    int lane = threadIdx.x & 31;
    const float* hp  = h   + (size_t)n * HC + hd * EMB;
    const float* asv = a_s + hd * EMB;
    const float* adv = a_d + hd * EMB;
    float ps = 0.f, pd = 0.f;
    for (int i = lane; i < EMB; i += 32) { float v = hp[i]; ps += v * asv[i]; pd += v * adv[i]; }
    for (int o = 16; o > 0; o >>= 1) { ps += __shfl_xor(ps, o, 32); pd += __shfl_xor(pd, o, 32); }
    if (lane == 0) { asn[n * 4 + hd] = ps; adn[n * 4 + hd] = pd; }
}

// ---------------------------------------------------------------------------
// CSR-by-dst build (once per timestep, reused across branches & layers)
// ---------------------------------------------------------------------------
__global__ void csr_count(const int* __restrict__ edges_t, int* __restrict__ cnt) {
    int e = blockIdx.x * blockDim.x + threadIdx.x;
    if (e >= ETOT) return;
    int dst = (e < NEDGES) ? edges_t[NEDGES + e] : (e - NEDGES);
    atomicAdd(&cnt[dst], 1);
}

__global__ void scan_excl(const int* __restrict__ cnt, int* __restrict__ off) {
    __shared__ int s[1024];
    const int tid = threadIdx.x;
    const int CH  = (NNODES + 1023) / 1024;
    int b0 = tid * CH, sum = 0;
    for (int i = 0; i < CH; ++i) { int idx = b0 + i; if (idx < NNODES) sum += cnt[idx]; }
    s[tid] = sum;
    __syncthreads();
    if (tid == 0) { int run = 0; for (int i = 0; i < 1024; ++i) { int v = s[i]; s[i] = run; run += v; } }
    __syncthreads();
    int run = s[tid];
    for (int i = 0; i < CH; ++i) { int idx = b0 + i; if (idx < NNODES) { off[idx] = run; run += cnt[idx]; } }
    if (tid == 1023) off[NNODES] = ETOT;
}

__global__ void csr_fill(const int* __restrict__ edges_t, int* __restrict__ cursor,
                         int* __restrict__ csr_src) {
    int e = blockIdx.x * blockDim.x + threadIdx.x;
    if (e >= ETOT) return;
    int src, dst;
    if (e < NEDGES) { src = edges_t[e]; dst = edges_t[NEDGES + e]; }
    else            { src = dst = e - NEDGES; }
    int pos = atomicAdd(&cursor[dst], 1);
    csr_src[pos] = src;
}

// ---------------------------------------------------------------------------
// Segment-softmax + neighborhood aggregation, atomic-free.
// Block = 128 threads = 4 waves; wave hd handles head hd of node blockIdx.x.
// Per edge: alpha computed once per wave, broadcast via shfl; 128 consecutive
// channels of h[src] loaded coalesced.
// ---------------------------------------------------------------------------
__global__ void gat_aggregate(const float* __restrict__ h,
                              const float* __restrict__ asn,
                              const float* __restrict__ adn,
                              const int*  __restrict__ off,
                              const int*  __restrict__ srcs,
                              float* __restrict__ out) {
    const int n    = blockIdx.x;
    const int hd   = threadIdx.x >> 5;
    const int lane = threadIdx.x & 31;
    const int base = off[n];
    const int deg  = off[n + 1] - base;
    const float adv = adn[n * 4 + hd];

    float m = -3.4e38f;
    for (int e = lane; e < deg; e += 32) {
        int s = srcs[base + e];
        float l = asn[s * 4 + hd] + adv;
        l = (l >= 0.f) ? l : 0.2f * l;
        m = fmaxf(m, l);
    }
    for (int o = 16; o > 0; o >>= 1) m = fmaxf(m, __shfl_xor(m, o, 32));

    float ssum = 0.f;
    for (int e = lane; e < deg; e += 32) {
        int s = srcs[base + e];
        float l = asn[s * 4 + hd] + adv;
        l = (l >= 0.f) ? l : 0.2f * l;
        ssum += __expf(l - m);
    }
    for (int o = 16; o > 0; o >>= 1) ssum += __shfl_xor(ssum, o, 32);
    const float inv = 1.0f / ssum;

    float a0 = 0.f, a1 = 0.f, a2 = 0.f, a3 = 0.f;
    for (int e0 = 0; e0 < deg; e0 += 32) {
        int e = e0 + lane;
        float alpha = 0.f; int sE = 0;
        if (e < deg) {
            sE = srcs[base + e];
            float l = asn[sE * 4 + hd] + adv;
            l = (l >= 0.f) ? l : 0.2f * l;
            alpha = __expf(l - m) * inv;
        }
        int cnt = min(32, deg - e0);
        for (int j = 0; j < cnt; ++j) {
            float aj = __shfl(alpha, j, 32);
            int   sj = __shfl(sE, j, 32);
            const float* hp = h + (size_t)sj * HC + hd * EMB + lane;
            a0 += aj * hp[0];  a1 += aj * hp[32];
            a2 += aj * hp[64]; a3 += aj * hp[96];
        }
    }
    float* op = out + (size_t)n * HC + hd * EMB + lane;
    op[0] = a0; op[32] = a1; op[64] = a2; op[96] = a3;
}

// bb[j] = lb[j] + sum_k b[k] * lW[j,k]   (fold GAT output bias into the linear)
__global__ void fused_bias(const float* __restrict__ b, const float* __restrict__ lW,
                           const float* __restrict__ lb, float* __restrict__ bb) {
    int j = threadIdx.x;
    float s = lb[j];
    const float* row = lW + (size_t)j * HC;
    for (int k = 0; k < HC; ++k) s += b[k] * row[k];
    bb[j] = s;
}

// Gather branch output at clf_nodes into reversed/concatenated flat buffer.
__global__ void gather_clf(const float* __restrict__ x, const int* __restrict__ clf,
                           float* __restrict__ flat, int colbase) {
    int b = blockIdx.x, c = threadIdx.x;
    flat[(size_t)b * FLATW + colbase + c] = x[(size_t)clf[b] * EMB + c];
}

// Tiny fc2: out[b, 0..1] = relu(hmid[b,:] . fc2_W[j,:] + fc2_b[j])
__global__ void fc2_kernel(const float* __restrict__ hm, const float* __restrict__ W,
                           const float* __restrict__ bias, float* __restrict__ out) {
    int b = blockIdx.x, lane = threadIdx.x;
    const float* hp = hm + (size_t)b * EMB;
    float p0 = 0.f, p1 = 0.f;
    for (int i = lane; i < EMB; i += 32) { float v = hp[i]; p0 += v * W[i]; p1 += v * W[EMB + i]; }
    for (int o = 16; o > 0; o >>= 1) { p0 += __shfl_xor(p0, o, 32); p1 += __shfl_xor(p1, o, 32); }
    if (lane == 0) {
        out[b * 2 + 0] = fmaxf(p0 + bias[0], 0.f);
        out[b * 2 + 1] = fmaxf(p1 + bias[1], 0.f);
    }
}

// ---------------------------------------------------------------------------
extern "C" void kernel_launch(void* const* d_in, const int* in_sizes, int n_in,
                              void* d_out, int out_size, void* d_ws, size_t ws_size,
                              hipStream_t stream) {
    const float* emb      = (const float*)d_in[0];   // [T,N,EMB]
    const int*   edges    = (const int*)  d_in[1];   // [T,2,E]
    const int*   clf      = (const int*)  d_in[5];   // [B]
    const float* gat_W[2]  = { (const float*)d_in[6],  (const float*)d_in[12] };
    const float* gat_as[2] = { (const float*)d_in[7],  (const float*)d_in[13] };
    const float* gat_ad[2] = { (const float*)d_in[8],  (const float*)d_in[14] };
    const float* gat_b[2]  = { (const float*)d_in[9],  (const float*)d_in[15] };
    const float* lin_W[2]  = { (const float*)d_in[10], (const float*)d_in[16] };
    const float* lin_b[2]  = { (const float*)d_in[11], (const float*)d_in[17] };
    const float* fc1_W = (const float*)d_in[18];
    const float* fc1_b = (const float*)d_in[19];
    const float* fc2_W = (const float*)d_in[20];
    const float* fc2_b = (const float*)d_in[21];
    float* out = (float*)d_out;

    // ---- workspace carve-up (≈150 MB) ----
    char* p = (char*)d_ws;
    auto takeF = [&](size_t n) { float* r = (float*)p; p += ((n * 4 + 255) & ~(size_t)255); return r; };
    auto takeI = [&](size_t n) { int*   r = (int*)p;   p += ((n * 4 + 255) & ~(size_t)255); return r; };
    float* hbuf = takeF((size_t)NNODES * HC);
    float* agg  = takeF((size_t)NNODES * HC);
    float* xb1  = takeF((size_t)NNODES * EMB);
    float* xb2  = takeF((size_t)NNODES * EMB);
    float* asn  = takeF((size_t)NNODES * HEADS);
    float* adn  = takeF((size_t)NNODES * HEADS);
    float* bb   = takeF(EMB);
    float* flat = takeF((size_t)BCLF * FLATW);
    float* hmid = takeF((size_t)BCLF * EMB);
    int* cnt    = takeI(NNODES);
    int* off    = takeI(NNODES + 1);
    int* cursor = takeI(NNODES);
    int* csrsrc = takeI(ETOT);

    const dim3 blk128(128), blk256(256);
    const int MBLK = (NNODES + 127) / 128;  // 157 (last block: 3 waves exit uniformly)

    for (int t = 0; t < T_WIN; ++t) {
        const int* edges_t = edges + (size_t)t * 2 * NEDGES;

        // ---- build dst-CSR for this timestep ----
        hipMemsetAsync(cnt, 0, NNODES * sizeof(int), stream);
        csr_count<<<dim3((ETOT + 255) / 256), blk256, 0, stream>>>(edges_t, cnt);
        scan_excl<<<dim3(1), dim3(1024), 0, stream>>>(cnt, off);
        hipMemcpyAsync(cursor, off, NNODES * sizeof(int), hipMemcpyDeviceToDevice, stream);
        csr_fill<<<dim3((ETOT + 255) / 256), blk256, 0, stream>>>(edges_t, cursor, csrsrc);

        for (int br = 0; br < 2; ++br) {
            const float* x = emb + (size_t)t * NNODES * EMB;
            for (int l = 0; l < LAYERS; ++l) {
                const float* W  = gat_W[br]  + (size_t)l * HC * EMB;
                const float* as = gat_as[br] + (size_t)l * HEADS * EMB;
                const float* ad = gat_ad[br] + (size_t)l * HEADS * EMB;
                const float* b  = gat_b[br]  + (size_t)l * HC;
                const float* lW = lin_W[br]  + (size_t)l * EMB * HC;
                const float* lb = lin_b[br]  + (size_t)l * EMB;
                float* xout = (l == 0) ? xb1 : xb2;

                fused_bias<<<dim3(1), blk128, 0, stream>>>(b, lW, lb, bb);
                // h = x @ W^T       [N,128]x[512,128]^T -> [N,512]
                gemm_wmma_f32<<<dim3(HC / 64, MBLK), blk128, 0, stream>>>(
                    x, W, nullptr, hbuf, NNODES, HC, EMB, 0);
                attn_coeff<<<dim3((NNODES * HEADS + 7) / 8), blk256, 0, stream>>>(
                    hbuf, as, ad, asn, adn);
                gat_aggregate<<<dim3(NNODES), blk128, 0, stream>>>(
                    hbuf, asn, adn, off, csrsrc, agg);
                // x' = relu(agg @ lW^T + bb)   [N,512]x[128,512]^T -> [N,128]
                gemm_wmma_f32<<<dim3(EMB / 64, MBLK), blk128, 0, stream>>>(
                    agg, lW, bb, xout, NNODES, EMB, HC, 1);
                x = xout;
            }
            // reversed timestep order, branch concat: cols (T-1-t)*256 + br*128
            gather_clf<<<dim3(BCLF), blk128, 0, stream>>>(
                x, clf, flat, (T_WIN - 1 - t) * (2 * EMB) + br * EMB);
        }
    }

    // head MLP: fc1 (WMMA) + fc2
    gemm_wmma_f32<<<dim3(EMB / 64, BCLF / 128), blk128, 0, stream>>>(
        flat, fc1_W, fc1_b, hmid, BCLF, EMB, FLATW, 1);
    fc2_kernel<<<dim3(BCLF), dim3(32), 0, stream>>>(hmid, fc2_W, fc2_b, out);
}